// Dispersion_3573412790706
// MI455X (gfx1250) — compile-verified
//
#include <hip/hip_runtime.h>
#include <math.h>

// ---------------- D3 constants ----------------
#define K1f      16.0f
#define K3f     (-4.0f)
#define CUTOFFf  10.0f
#define NZi      95

// 4-float vector with alignment relaxed to 4 bytes: gfx1250 supports
// unaligned VMEM, so loads of this type emit global_load_b128 even though the
// 300-byte (75-float) c6ab blocks are only 4-byte aligned.
typedef float f4v __attribute__((ext_vector_type(4)));
typedef f4v f4a __attribute__((aligned(4)));

// Wave32 segmented sum over contiguous runs of equal `seg` (idx_i is sorted).
// Run-head position is found once with ballot+clz; the 5-level scan then only
// shuffles the float value (5 ds_bpermute instead of 10), and one more shuffle
// finds run tails. Tail lanes commit with a single f32 atomic (~2 per wave).
__device__ __forceinline__ void seg_commit(int seg, float v, unsigned lane,
                                           float* __restrict__ acc) {
  int up = __shfl_up(seg, 1u, 32);
  bool head = (lane == 0u) || (up != seg);
  unsigned hmask = (unsigned)__ballot(head);          // wave32: low 32 bits
  // Highest set head bit at position <= lane == this lane's run start.
  // (hmask << (31-lane)) puts bit `lane` at bit 31; clz = lane - start.
  unsigned start = lane - (unsigned)__clz(hmask << (31u - lane));
#pragma unroll
  for (unsigned off = 1; off < 32; off <<= 1) {
    float u = __shfl_up(v, off, 32);
    if (lane >= start + off) v += u;                  // stay within the run
  }
  int dn = __shfl_down(seg, 1u, 32);
  bool tail = (lane == 31u) || (dn != seg);
  if (tail && seg >= 0) {
    unsafeAtomicAdd(&acc[seg], v);                    // global_atomic_add_f32
  }
}

__global__ __launch_bounds__(256) void d3_zero_k(float* nc, float* out, int n) {
  int i = blockIdx.x * blockDim.x + threadIdx.x;
  if (i < n) { nc[i] = 0.0f; out[i] = 0.0f; }
}

// ---- Phase 1: fractional coordination numbers nc[i] = seg_sum(damp) --------
__global__ __launch_bounds__(256) void d3_cn_k(
    const int* __restrict__ Z, const float* __restrict__ Dij,
    const int* __restrict__ idx_i, const int* __restrict__ idx_j,
    const float* __restrict__ rcov, float* __restrict__ nc, int E) {
  int e = blockIdx.x * blockDim.x + threadIdx.x;
  unsigned lane = threadIdx.x & 31u;
  int seg = -1;
  float damp = 0.0f;
  if (e < E) {
    int i = idx_i[e], j = idx_j[e];
    int Zi = Z[i], Zj = Z[j];
    float r = Dij[e];
    float rco = rcov[Zi] + rcov[Zj];
    float t = K1f * (__fdividef(rco, r) - 1.0f);
    float sig = __fdividef(1.0f, 1.0f + __expf(-t));
    // smootherstep with (cutoff - cuton) == 1
    float x = fminf(fmaxf(CUTOFFf - r, 0.0f), 1.0f);
    float sw = x * x * x * (x * (6.0f * x - 15.0f) + 10.0f);
    damp = sig * sw;
    seg = i;
  }
  seg_commit(seg, damp, lane, nc);
}

// ---- Phase 2: C6 interpolation + BJ-damped energy, seg_sum(epair) ----------
__global__ __launch_bounds__(256) void d3_edisp_k(
    const int* __restrict__ Z, const float* __restrict__ Dij,
    const int* __restrict__ idx_i, const int* __restrict__ idx_j,
    const float* __restrict__ c6ab, const float* __restrict__ nc,
    const float* __restrict__ r2r4,
    const float* __restrict__ s6p, const float* __restrict__ s8p,
    const float* __restrict__ a1p, const float* __restrict__ a2p,
    float* __restrict__ out, int E) {
  int e = blockIdx.x * blockDim.x + threadIdx.x;
  unsigned lane = threadIdx.x & 31u;
  int seg = -1;
  float epair = 0.0f;
  if (e < E) {
    int i = idx_i[e], j = idx_j[e];
    int Zi = Z[i], Zj = Z[j];
    const float* pars = c6ab + (size_t)(Zi * NZi + Zj) * 75u;
    // gfx1250 global_prefetch_b8: pull the 300B block toward the WGP while the
    // nc/Dij/r2r4 loads are still in flight.
    __builtin_prefetch(pars,      0, 1);
    __builtin_prefetch(pars + 32, 0, 1);
    __builtin_prefetch(pars + 64, 0, 1);
    float nci = nc[i], ncj = nc[j];
    float r = Dij[e];

    float rsum = 0.0f, csum = 0.0f;
    float dmin = 3.4e38f, c6mem = 0.0f;

    auto accum = [&](float cn0, float cn1, float cn2) {
      float d1 = cn1 - nci;
      float d2 = cn2 - ncj;
      float dist = fmaf(d1, d1, d2 * d2);
      float w = __expf(K3f * dist);          // v_exp_f32
      if (cn0 > 0.0f) {                      // -> branchless v_cndmask adds
        rsum += w;
        csum += w * cn0;
      }
      if (dist < dmin) { dmin = dist; c6mem = cn0; }
    };

    // 24 triples via 18 underaligned b128 loads (4 triples per 3 loads)
    const f4a* p4 = (const f4a*)pars;
#pragma unroll
    for (int g = 0; g < 6; ++g) {
      f4v q0 = p4[3 * g + 0];
      f4v q1 = p4[3 * g + 1];
      f4v q2 = p4[3 * g + 2];
      accum(q0.x, q0.y, q0.z);
      accum(q0.w, q1.x, q1.y);
      accum(q1.z, q1.w, q2.x);
      accum(q2.y, q2.z, q2.w);
    }
    // 25th triple (contiguous scalars -> merged b96; avoids OOB past table)
    accum(pars[72], pars[73], pars[74]);

    float c6 = (rsum > 0.0f) ? __fdividef(csum, rsum) : c6mem;
    float t3 = 3.0f * r2r4[Zi] * r2r4[Zj];   // == c8/c6 exactly
    float c8 = c6 * t3;

    float a1 = *a1p, a2 = *a2p, s6 = *s6p, s8 = *s8p;  // uniform s_loads
    float tmp  = fmaf(a1, __fsqrt_rn(t3), a2);
    float r2   = r * r;
    float r6   = r2 * r2 * r2;
    float r8   = r6 * r2;
    float tmp2 = tmp * tmp;
    float tmp6 = tmp2 * tmp2 * tmp2;
    float tmp8 = tmp6 * tmp2;
    const float cut6 = 1.0e6f, cut8 = 1.0e8f;
    float c6t = cut6 + tmp6;
    float c8t = cut8 + tmp8;
    float shift = r * 0.1f - 1.0f;           // Dij/CUTOFF - 1
    float e6 = __fdividef(1.0f, r6 + tmp6) - __fdividef(1.0f, c6t)
             + 6.0f * cut6 * __fdividef(shift, c6t * c6t);
    float e8 = __fdividef(1.0f, r8 + tmp8) - __fdividef(1.0f, c8t)
             + 8.0f * cut8 * __fdividef(shift, c8t * c8t);
    if (r >= CUTOFFf) { e6 = 0.0f; e8 = 0.0f; }
    epair = -0.5f * (s6 * c6 * e6 + s8 * c8 * e8);
    seg = i;
  }
  seg_commit(seg, epair, lane, out);
}

extern "C" void kernel_launch(void* const* d_in, const int* in_sizes, int n_in,
                              void* d_out, int out_size, void* d_ws, size_t ws_size,
                              hipStream_t stream) {
  const int*   Z     = (const int*)  d_in[0];
  const float* Dij   = (const float*)d_in[1];
  const int*   idx_i = (const int*)  d_in[2];
  const int*   idx_j = (const int*)  d_in[3];
  const float* c6ab  = (const float*)d_in[4];
  const float* rcov  = (const float*)d_in[5];
  const float* r2r4  = (const float*)d_in[6];
  const float* s6    = (const float*)d_in[7];
  const float* s8    = (const float*)d_in[8];
  const float* a1    = (const float*)d_in[9];
  const float* a2    = (const float*)d_in[10];

  const int N = in_sizes[0];
  const int E = in_sizes[1];
  float* out = (float*)d_out;
  float* nc  = (float*)d_ws;          // N floats of scratch (128 KB)

  const int TB = 256;                 // 8 wave32 waves per block
  d3_zero_k<<<(N + TB - 1) / TB, TB, 0, stream>>>(nc, out, N);
  d3_cn_k<<<(E + TB - 1) / TB, TB, 0, stream>>>(Z, Dij, idx_i, idx_j, rcov, nc, E);
  d3_edisp_k<<<(E + TB - 1) / TB, TB, 0, stream>>>(Z, Dij, idx_i, idx_j, c6ab, nc,
                                                   r2r4, s6, s8, a1, a2, out, E);
}